// PyramidalLK_23716809409206
// MI455X (gfx1250) — compile-verified
//
#include <hip/hip_runtime.h>
#include <math.h>

// CDNA5 / gfx1250 pyramidal Lucas-Kanade optical flow.
// Bandwidth-bound pipeline; the 5x5 box filter is executed on the FP32 matrix
// pipe as banded-matrix WMMAs (v_wmma_f32_16x16x4_f32), one wave per 16x16 tile.
// Band-matrix operand fragments are hoisted into registers (computed once per
// wave) so the WMMA stream is not gated by per-op VALU coefficient rebuilds.

typedef __attribute__((ext_vector_type(2))) float v2f;
typedef __attribute__((ext_vector_type(8))) float v8f;

#define NLVL 4

// ---------------------------------------------------------------- grayscale
__global__ __launch_bounds__(256)
void k_gray(const float* __restrict__ img, float* __restrict__ gray,
            int B, int H, int W) {
  int idx = blockIdx.x * blockDim.x + threadIdx.x;
  int hw = H * W;
  int total = B * hw;
  if (idx >= total) return;
  int b = idx / hw;
  int p = idx - b * hw;
  const float* s = img + (size_t)b * 3 * hw;
  gray[idx] = (s[p] + s[p + hw] + s[p + 2 * hw]) * (1.0f / 3.0f);
}

// ---------------------------------------------------------------- 2x2 avg pool
__global__ __launch_bounds__(256)
void k_down(const float* __restrict__ src, float* __restrict__ dst,
            int B, int Hd, int Wd) {
  int idx = blockIdx.x * blockDim.x + threadIdx.x;
  int hw = Hd * Wd;
  int total = B * hw;
  if (idx >= total) return;
  int b = idx / hw;
  int p = idx - b * hw;
  int r = p / Wd;
  int c = p - r * Wd;
  int Hs = Hd * 2, Ws = Wd * 2;
  const float* s = src + (size_t)b * Hs * Ws;
  int r2 = 2 * r, c2 = 2 * c;
  dst[idx] = 0.25f * (s[r2 * Ws + c2] + s[r2 * Ws + c2 + 1] +
                      s[(r2 + 1) * Ws + c2] + s[(r2 + 1) * Ws + c2 + 1]);
}

// ---------------------------------------------------------------- zero fill
__global__ __launch_bounds__(256)
void k_zero(float* __restrict__ p, int n) {
  int idx = blockIdx.x * blockDim.x + threadIdx.x;
  if (idx < n) p[idx] = 0.0f;
}

// ------------------------------------------------ bilinear x2 upsample (ac), *2
__global__ __launch_bounds__(256)
void k_up2(const float* __restrict__ fin, float* __restrict__ fout,
           int B, int Hs, int Ws) {
  int Hd = Hs * 2, Wd = Ws * 2;
  int idx = blockIdx.x * blockDim.x + threadIdx.x;
  int total = B * 2 * Hd * Wd;
  if (idx >= total) return;
  int c = idx % Wd;
  int r = (idx / Wd) % Hd;
  int ch = (idx / (Wd * Hd)) % 2;
  int b = idx / (Wd * Hd * 2);
  float sy = (float)(Hs - 1) / (float)(Hd - 1);
  float sx = (float)(Ws - 1) / (float)(Wd - 1);
  float ys = r * sy, xs = c * sx;
  int y0 = (int)floorf(ys);
  int x0 = (int)floorf(xs);
  if (y0 < 0) y0 = 0; if (y0 > Hs - 2) y0 = Hs - 2;
  if (x0 < 0) x0 = 0; if (x0 > Ws - 2) x0 = Ws - 2;
  float wy = ys - (float)y0;
  float wx = xs - (float)x0;
  const float* s = fin + ((size_t)(b * 2 + ch) * Hs) * Ws;
  float v00 = s[y0 * Ws + x0];
  float v01 = s[y0 * Ws + x0 + 1];
  float v10 = s[(y0 + 1) * Ws + x0];
  float v11 = s[(y0 + 1) * Ws + x0 + 1];
  float val = v00 * (1.f - wx) * (1.f - wy) + v01 * wx * (1.f - wy) +
              v10 * (1.f - wx) * wy + v11 * wx * wy;
  fout[((size_t)(b * 2 + ch) * Hd + r) * Wd + c] = 2.0f * val;
}

// --------------------- fused: warp I2 by flow + Sobel(I1) + 5 tensor products
__global__ __launch_bounds__(256)
void k_warp_stats(const float* __restrict__ g1, const float* __restrict__ g2,
                  const float* __restrict__ flow, float* __restrict__ stats,
                  int B, int H, int W) {
  int idx = blockIdx.x * blockDim.x + threadIdx.x;
  int hw = H * W;
  int total = B * hw;
  if (idx >= total) return;
  int b = idx / hw;
  int p = idx - b * hw;
  int r = p / W;
  int c = p - r * W;
  const float* G1 = g1 + (size_t)b * hw;
  const float* G2 = g2 + (size_t)b * hw;

  auto g1at = [&](int rr, int cc) -> float {
    return (rr >= 0 && rr < H && cc >= 0 && cc < W) ? G1[rr * W + cc] : 0.0f;
  };
  // SX = [[-1,0,1],[-2,0,2],[-1,0,1]]/8 cross-correlation, zero padded
  float ix = (-g1at(r - 1, c - 1) + g1at(r - 1, c + 1)
              - 2.f * g1at(r, c - 1) + 2.f * g1at(r, c + 1)
              - g1at(r + 1, c - 1) + g1at(r + 1, c + 1)) * 0.125f;
  float iy = (-g1at(r - 1, c - 1) - 2.f * g1at(r - 1, c) - g1at(r - 1, c + 1)
              + g1at(r + 1, c - 1) + 2.f * g1at(r + 1, c) + g1at(r + 1, c + 1)) * 0.125f;

  float u = flow[((size_t)(b * 2 + 0) * H + r) * W + c];
  float v = flow[((size_t)(b * 2 + 1) * H + r) * W + c];
  float x = (float)c + u;
  float y = (float)r + v;
  float x0f = floorf(x), y0f = floorf(y);
  float wx = x - x0f, wy = y - y0f;
  int x0 = (int)x0f, y0 = (int)y0f;
  auto samp = [&](int yy, int xx) -> float {
    bool valid = (xx >= 0 && xx <= W - 1 && yy >= 0 && yy <= H - 1);
    int xc = xx < 0 ? 0 : (xx > W - 1 ? W - 1 : xx);
    int yc = yy < 0 ? 0 : (yy > H - 1 ? H - 1 : yy);
    return valid ? G2[yc * W + xc] : 0.0f;
  };
  float i2w = samp(y0, x0) * (1.f - wx) * (1.f - wy)
            + samp(y0, x0 + 1) * wx * (1.f - wy)
            + samp(y0 + 1, x0) * (1.f - wx) * wy
            + samp(y0 + 1, x0 + 1) * wx * wy;
  float it = i2w - G1[r * W + c];

  size_t ps = (size_t)hw;
  float* sp = stats + (size_t)b * 5 * ps + (size_t)r * W + c;
  sp[0]      = ix * ix;
  sp[ps]     = iy * iy;
  sp[2 * ps] = ix * iy;
  sp[3 * ps] = ix * it;
  sp[4 * ps] = iy * it;
}

// ---- 5x5 box filter via banded-matrix WMMAs + 2x2 LK solve + flow update ----
// One wave per 16x16 output tile. Band matrices:
//   band0(m,k) = 1 iff k in [m, m+4]      (top 16x16 block of the 5-tap band)
//   band1(m,k) = 1 iff m in [k+12, k+16]  (halo block; only k<4 is nonzero)
// Identity used: the horizontal-pass B fragment H0[K][N] = band0(N,K) evaluates
// per-lane to the same value as the vertical-pass A fragment band0(ln,K), so a
// single register-resident set of fragments serves both passes.
__device__ __forceinline__ float band0f(int m, int k) {
  return (k >= m && k <= m + 4) ? 1.0f : 0.0f;
}
__device__ __forceinline__ float band1f(int m, int k) {
  return (m >= k + 12 && m <= k + 16) ? 1.0f : 0.0f;
}

__global__ __launch_bounds__(32)
void k_box_solve(const float* __restrict__ stats,
                 const float* __restrict__ flow_in,
                 float* __restrict__ flow_out,
                 int B, int H, int W) {
  __shared__ float P[5][20][32];  // stats halo tile: row r = imgRow tile*16+r-2
  __shared__ float T[16][32];     // vertical-pass result (per plane, reused)

  const int lane = threadIdx.x;       // 0..31, one wave
  const int tx = blockIdx.x, ty = blockIdx.y, b = blockIdx.z;
  const int hi = lane >> 4;           // lane half
  const int ln = lane & 15;

  // ---- band fragments: computed ONCE, reused by all 75 WMMAs ----
  v2f bnd0[4], bnd1;
  #pragma unroll
  for (int kc = 0; kc < 4; ++kc) {
    bnd0[kc].x = band0f(ln, kc * 4 + 2 * hi + 0);
    bnd0[kc].y = band0f(ln, kc * 4 + 2 * hi + 1);
  }
  bnd1.x = band1f(ln, 2 * hi + 0);
  bnd1.y = band1f(ln, 2 * hi + 1);

  // ---- load 20x20 halo tile of all 5 planes (zero-pad outside image & c>=20)
  {
    const int c = lane;  // LDS column
    for (int p = 0; p < 5; ++p) {
      const float* sp = stats + ((size_t)b * 5 + p) * H * W;
      // prefetch this plane's tile rows (L2 is 192MB: set resident, hide miss)
      {
        int ir = ty * 16 - 2 + (lane & 3) * 6;
        int ic = tx * 16 - 2 + (lane >> 2) * 4;
        if (ir >= 0 && ir < H && ic >= 0 && ic < W)
          __builtin_prefetch(sp + ir * W + ic, 0, 3);
      }
      for (int r = 0; r < 20; ++r) {
        float val = 0.0f;
        if (c < 20) {
          int ir = ty * 16 + r - 2;
          int ic = tx * 16 + c - 2;
          if (ir >= 0 && ir < H && ic >= 0 && ic < W) val = sp[ir * W + ic];
        }
        P[p][r][c] = val;
      }
    }
  }
  __syncthreads();

  v8f S[5];
  for (int p = 0; p < 5; ++p) {
    // ---- vertical pass: T = band * P, for two 16-column tiles
    #pragma unroll
    for (int t = 0; t < 2; ++t) {
      v8f acc = {};
      #pragma unroll
      for (int kc = 0; kc < 4; ++kc) {   // top block, K chunks 0..3
        v2f bm;
        bm.x = P[p][kc * 4 + 2 * hi + 0][t * 16 + ln];
        bm.y = P[p][kc * 4 + 2 * hi + 1][t * 16 + ln];
        acc = __builtin_amdgcn_wmma_f32_16x16x4_f32(false, bnd0[kc], false, bm,
                                                    (short)0, acc, false, false);
      }
      {  // bottom halo block: only K chunk 0 carries nonzero band coefficients
        v2f bm;
        bm.x = P[p][16 + 2 * hi + 0][t * 16 + ln];
        bm.y = P[p][16 + 2 * hi + 1][t * 16 + ln];
        acc = __builtin_amdgcn_wmma_f32_16x16x4_f32(false, bnd1, false, bm,
                                                    (short)0, acc, false, false);
      }
      // D layout: VGPR v -> row v+8*hi, col = t*16 + ln
      #pragma unroll
      for (int v = 0; v < 8; ++v) T[v + 8 * hi][t * 16 + ln] = acc[v];
    }
    __syncthreads();

    // ---- horizontal pass: Out = T * band^T (T0 full, T1 halo chunk only)
    v8f acc = {};
    #pragma unroll
    for (int kc = 0; kc < 4; ++kc) {
      v2f a;
      a.x = T[ln][kc * 4 + 2 * hi + 0];
      a.y = T[ln][kc * 4 + 2 * hi + 1];
      acc = __builtin_amdgcn_wmma_f32_16x16x4_f32(false, a, false, bnd0[kc],
                                                  (short)0, acc, false, false);
    }
    {
      v2f a;
      a.x = T[ln][16 + 2 * hi + 0];
      a.y = T[ln][16 + 2 * hi + 1];
      acc = __builtin_amdgcn_wmma_f32_16x16x4_f32(false, a, false, bnd1,
                                                  (short)0, acc, false, false);
    }
    S[p] = acc;
    __syncthreads();  // T reused next plane (WAR)
  }

  // ---- 2x2 LK solve + flow update (H,W multiples of 16 -> no bounds checks)
  #pragma unroll
  for (int v = 0; v < 8; ++v) {
    int row = ty * 16 + v + 8 * hi;
    int col = tx * 16 + ln;
    float sxx = S[0][v], syy = S[1][v], sxy = S[2][v];
    float sxt = S[3][v], syt = S[4][v];
    float inv = 1.0f / (sxx * syy - sxy * sxy + 1e-6f);
    float du = (-syy * sxt + sxy * syt) * inv;
    float dv = ( sxy * sxt - sxx * syt) * inv;
    size_t i0 = ((size_t)(b * 2 + 0) * H + row) * W + col;
    size_t i1 = ((size_t)(b * 2 + 1) * H + row) * W + col;
    flow_out[i0] = flow_in[i0] + du;
    flow_out[i1] = flow_in[i1] + dv;
  }
}

// ------------------------------------------------------------------ driver
extern "C" void kernel_launch(void* const* d_in, const int* in_sizes, int n_in,
                              void* d_out, int out_size, void* d_ws, size_t ws_size,
                              hipStream_t stream) {
  (void)in_sizes; (void)n_in; (void)out_size; (void)ws_size;
  const float* img1 = (const float*)d_in[0];
  const float* img2 = (const float*)d_in[1];
  const int B = 8, H0 = 512, W0 = 512;

  int Hl[NLVL], Wl[NLVL];
  size_t lvlOff[NLVL];
  size_t pyr = 0;
  for (int l = 0; l < NLVL; ++l) {
    Hl[l] = H0 >> l;
    Wl[l] = W0 >> l;
    lvlOff[l] = pyr;
    pyr += (size_t)B * Hl[l] * Wl[l];
  }

  float* ws = (float*)d_ws;
  float* g1 = ws;                                       // gray pyramid img1
  float* g2 = g1 + pyr;                                 // gray pyramid img2
  float* fA = g2 + pyr;                                 // flow buffer A
  float* fB = fA + (size_t)B * 2 * H0 * W0;             // flow buffer B
  float* st = fB + (size_t)B * 2 * H0 * W0;             // 5 stats planes

  auto nb = [](int n) { return (n + 255) / 256; };

  // grayscale at full res, then grayscale pyramid (commutes with mean/pool/warp)
  {
    int n = B * H0 * W0;
    k_gray<<<nb(n), 256, 0, stream>>>(img1, g1 + lvlOff[0], B, H0, W0);
    k_gray<<<nb(n), 256, 0, stream>>>(img2, g2 + lvlOff[0], B, H0, W0);
  }
  for (int l = 1; l < NLVL; ++l) {
    int n = B * Hl[l] * Wl[l];
    k_down<<<nb(n), 256, 0, stream>>>(g1 + lvlOff[l - 1], g1 + lvlOff[l], B, Hl[l], Wl[l]);
    k_down<<<nb(n), 256, 0, stream>>>(g2 + lvlOff[l - 1], g2 + lvlOff[l], B, Hl[l], Wl[l]);
  }

  // coarsest flow = 0
  {
    int n = B * 2 * Hl[NLVL - 1] * Wl[NLVL - 1];
    k_zero<<<nb(n), 256, 0, stream>>>(fA, n);
  }

  float* fcur = fA;
  float* fother = fB;
  for (int l = NLVL - 1; l >= 0; --l) {
    int H = Hl[l], W = Wl[l];
    if (l < NLVL - 1) {
      int n = B * 2 * H * W;
      k_up2<<<nb(n), 256, 0, stream>>>(fcur, fother, B, Hl[l + 1], Wl[l + 1]);
      float* t = fcur; fcur = fother; fother = t;
    }
    int n = B * H * W;
    k_warp_stats<<<nb(n), 256, 0, stream>>>(g1 + lvlOff[l], g2 + lvlOff[l],
                                            fcur, st, B, H, W);
    float* fout = (l == 0) ? (float*)d_out : fcur;
    dim3 grid(W / 16, H / 16, B);
    k_box_solve<<<grid, 32, 0, stream>>>(st, fcur, fout, B, H, W);
  }
}